// GAT_40407052320898
// MI455X (gfx1250) — compile-verified
//
#include <hip/hip_runtime.h>

// ---------------------------------------------------------------------------
// 3-layer GAT forward for MI455X (gfx1250, wave32).
//   - GEMMs  : V_WMMA_F32_16X16X4_F32 (exact fp32 matrix path)
//   - softmax: float->uint monotone key + atomicMax(u32), then exp-sum
//   - scatter: wave-per-edge, float4 per lane, global f32 atomics (L2 RMW)
// ---------------------------------------------------------------------------

typedef float v2f __attribute__((ext_vector_type(2)));
typedef float v8f __attribute__((ext_vector_type(8)));

#define HD 128  // hidden width of every layer

// -------- monotone float <-> uint key (for atomicMax-based segment max) ----
__device__ __forceinline__ unsigned gat_f2key(float x) {
  unsigned u = __float_as_uint(x);
  return (u & 0x80000000u) ? ~u : (u | 0x80000000u);
}
__device__ __forceinline__ float gat_key2f(unsigned k) {
  return __uint_as_float((k & 0x80000000u) ? (k & 0x7FFFFFFFu) : ~k);
}
__device__ __forceinline__ float gat_lrelu(float v) {
  return v > 0.0f ? v : 0.2f * v;
}
// edge e in [0, E) comes from edge_index; e in [E, E+N) is self-loop e-E.
__device__ __forceinline__ void gat_edge(const int* __restrict__ ei, long long E,
                                         long long e, int& s, int& d) {
  if (e < E) { s = ei[e]; d = ei[E + e]; }
  else       { s = d = (int)(e - E); }
}

// -------- h = A[N,K] @ W[K,128] : one wave per 16x16 output tile -----------
__global__ void gat_gemm_wmma(const float* __restrict__ A,
                              const float* __restrict__ W,
                              float* __restrict__ C, int N, int K) {
  const int lane = threadIdx.x & 31;
  const long long wave = (((long long)blockIdx.x * blockDim.x) + threadIdx.x) >> 5;
  const int tilesN = HD / 16;                    // 8 tiles along the 128 dim
  const long long tm = wave / tilesN;            // wave-uniform
  const int       tn = (int)(wave % tilesN);
  if (tm * 16 >= (long long)N) return;           // uniform across the wave

  // A fragment: lanes 0-15 hold {K=0,K=1}, lanes 16-31 hold {K=2,K=3}
  const int mrow = (int)tm * 16 + (lane & 15);
  const int ksub = (lane >> 4) * 2;
  // B fragment: lane = column n, VGPR pair = {K=ksub, K=ksub+1}
  const int ncol = tn * 16 + (lane & 15);

  const float* arow = A + (long long)mrow * K;
  v8f acc = {};
  for (int k = 0; k < K; k += 4) {
    v2f a, b;
    a.x = arow[k + ksub];
    a.y = arow[k + ksub + 1];
    b.x = W[(long long)(k + ksub)     * HD + ncol];
    b.y = W[(long long)(k + ksub + 1) * HD + ncol];
    acc = __builtin_amdgcn_wmma_f32_16x16x4_f32(false, a, false, b,
                                                (short)0, acc, false, false);
  }
  // C/D layout: VGPR r -> row tm*16 + r + 8*(lane>=16), col = lane&15
  const int row0 = (int)tm * 16 + (lane >> 4) * 8;
#pragma unroll
  for (int r = 0; r < 8; ++r)
    C[(long long)(row0 + r) * HD + ncol] = acc[r];
}

// -------- per-node attention logits + per-layer state init -----------------
__global__ void gat_alpha_init(const float* __restrict__ h,
                               const float* __restrict__ a_src,
                               const float* __restrict__ a_dst,
                               float* __restrict__ as_, float* __restrict__ ad_,
                               unsigned* __restrict__ emaxKey,
                               float* __restrict__ denom, int N) {
  const int lane = threadIdx.x & 31;
  const long long node = (((long long)blockIdx.x * blockDim.x) + threadIdx.x) >> 5;
  if (node >= (long long)N) return;
  const float* hp = h + node * (long long)HD;
  float ss = 0.0f, sd = 0.0f;
#pragma unroll
  for (int i = 0; i < HD; i += 32) {
    float v = hp[i + lane];
    ss += v * a_src[i + lane];
    sd += v * a_dst[i + lane];
  }
#pragma unroll
  for (int off = 16; off > 0; off >>= 1) {
    ss += __shfl_xor(ss, off, 32);
    sd += __shfl_xor(sd, off, 32);
  }
  if (lane == 0) {
    as_[node]     = ss;
    ad_[node]     = sd;
    emaxKey[node] = 0x007FFFFFu;   // key(-inf)
    denom[node]   = 0.0f;
  }
}

// -------- pass 1: segment max over dst (thread per edge) -------------------
__global__ void gat_edge_max(const int* __restrict__ ei, long long E, long long ET,
                             const float* __restrict__ as_,
                             const float* __restrict__ ad_,
                             unsigned* __restrict__ emaxKey) {
  long long e = (long long)blockIdx.x * blockDim.x + threadIdx.x;
  if (e >= ET) return;
  int s, d; gat_edge(ei, E, e, s, d);
  float v = gat_lrelu(as_[s] + ad_[d]);
  atomicMax(&emaxKey[d], gat_f2key(v));
}

// -------- pass 2: exp-sum denominator (thread per edge) --------------------
__global__ void gat_edge_expsum(const int* __restrict__ ei, long long E, long long ET,
                                const float* __restrict__ as_,
                                const float* __restrict__ ad_,
                                const unsigned* __restrict__ emaxKey,
                                float* __restrict__ denom) {
  long long e = (long long)blockIdx.x * blockDim.x + threadIdx.x;
  if (e >= ET) return;
  int s, d; gat_edge(ei, E, e, s, d);
  float v = gat_lrelu(as_[s] + ad_[d]);
  atomicAdd(&denom[d], __expf(v - gat_key2f(emaxKey[d])));
}

// -------- pass 3: weighted scatter (wave per edge, float4 per lane) --------
__global__ void gat_edge_scatter(const int* __restrict__ ei, long long E, long long ET,
                                 const float* __restrict__ as_,
                                 const float* __restrict__ ad_,
                                 const unsigned* __restrict__ emaxKey,
                                 const float* __restrict__ denom,
                                 const float* __restrict__ h,
                                 float* __restrict__ out) {
  const int lane = threadIdx.x & 31;
  const long long e = (((long long)blockIdx.x * blockDim.x) + threadIdx.x) >> 5;
  if (e >= ET) return;                                   // wave-uniform
  int s, d; gat_edge(ei, E, e, s, d);
  float v = gat_lrelu(as_[s] + ad_[d]);
  float w = __expf(v - gat_key2f(emaxKey[d])) / denom[d];

  const float4 hv = ((const float4*)(h + (long long)s * HD))[lane];
  float* op = out + (long long)d * HD + lane * 4;
  atomicAdd(op + 0, w * hv.x);
  atomicAdd(op + 1, w * hv.y);
  atomicAdd(op + 2, w * hv.z);
  atomicAdd(op + 3, w * hv.w);
}

// -------- bias (+ optional ReLU) -------------------------------------------
__global__ void gat_bias_act(float* __restrict__ out, const float* __restrict__ b,
                             long long total, int relu) {
  long long i = (long long)blockIdx.x * blockDim.x + threadIdx.x;
  if (i >= total) return;
  float v = out[i] + b[(int)(i & (HD - 1))];
  out[i] = relu ? fmaxf(v, 0.0f) : v;
}

// -------- readout: d_out = mean_n(h)@Wp + bp -------------------------------
__global__ void gat_set_out(float* d_out, const float* __restrict__ bp) {
  if (threadIdx.x == 0 && blockIdx.x == 0) d_out[0] = bp[0];
}
__global__ void gat_readout(const float* __restrict__ h,
                            const float* __restrict__ Wp,
                            float* __restrict__ d_out, int N) {
  const int f = blockIdx.x;          // one block per feature, HD blocks
  float sum = 0.0f;
  for (int n = threadIdx.x; n < N; n += blockDim.x)
    sum += h[(long long)n * HD + f];
  __shared__ float sm[256];
  sm[threadIdx.x] = sum;
  __syncthreads();
  for (int s = 128; s > 0; s >>= 1) {
    if (threadIdx.x < s) sm[threadIdx.x] += sm[threadIdx.x + s];
    __syncthreads();
  }
  if (threadIdx.x == 0)
    atomicAdd(d_out, sm[0] * (1.0f / (float)N) * Wp[f]);
}

// ---------------------------------------------------------------------------
extern "C" void kernel_launch(void* const* d_in, const int* in_sizes, int n_in,
                              void* d_out, int out_size, void* d_ws, size_t ws_size,
                              hipStream_t stream) {
  const float* x  = (const float*)d_in[0];
  const int*   ei = (const int*)d_in[1];
  const float* Wl[3]  = {(const float*)d_in[2],  (const float*)d_in[3],  (const float*)d_in[4]};
  const float* asl[3] = {(const float*)d_in[5],  (const float*)d_in[6],  (const float*)d_in[7]};
  const float* adl[3] = {(const float*)d_in[8],  (const float*)d_in[9],  (const float*)d_in[10]};
  const float* bl[3]  = {(const float*)d_in[11], (const float*)d_in[12], (const float*)d_in[13]};
  const float* Wp = (const float*)d_in[14];
  const float* bp = (const float*)d_in[15];

  const int       N  = in_sizes[0] / 64;       // 100000
  const long long E  = in_sizes[1] / 2;        // 1600000
  const long long ET = E + (long long)N;       // + self loops

  // workspace layout (floats)
  float*    hA       = (float*)d_ws;                      // [N, 128] gemm out
  float*    hB       = hA + (long long)N * HD;            // [N, 128] aggregate
  float*    as_      = hB + (long long)N * HD;            // [N]
  float*    ad_      = as_ + N;                           // [N]
  unsigned* emaxKey  = (unsigned*)(ad_ + N);              // [N]
  float*    denom    = (float*)(emaxKey + N);             // [N]

  const float* cur = x;
  int curK = 64;
  for (int L = 0; L < 3; ++L) {
    // 1) h = cur @ W_L  (one wave per 16x16 tile)
    const long long waves = (long long)(N / 16) * (HD / 16);
    gat_gemm_wmma<<<(unsigned)((waves * 32 + 255) / 256), 256, 0, stream>>>(
        cur, Wl[L], hA, N, curK);
    // 2) logits + per-layer state init (one wave per node)
    gat_alpha_init<<<(unsigned)(((long long)N * 32 + 255) / 256), 256, 0, stream>>>(
        hA, asl[L], adl[L], as_, ad_, emaxKey, denom, N);
    // 3) zero the aggregation buffer (graph-capturable memset node)
    hipMemsetAsync(hB, 0, (size_t)N * HD * sizeof(float), stream);
    // 4) segment softmax (max, then exp-sum)
    const unsigned tb = (unsigned)((ET + 255) / 256);
    gat_edge_max<<<tb, 256, 0, stream>>>(ei, E, ET, as_, ad_, emaxKey);
    gat_edge_expsum<<<tb, 256, 0, stream>>>(ei, E, ET, as_, ad_, emaxKey, denom);
    // 5) weighted scatter (wave per edge)
    gat_edge_scatter<<<(unsigned)((ET * 32 + 255) / 256), 256, 0, stream>>>(
        ei, E, ET, as_, ad_, emaxKey, denom, hA, hB);
    // 6) bias (+ReLU on layers 0,1)
    gat_bias_act<<<(unsigned)(((long long)N * HD + 255) / 256), 256, 0, stream>>>(
        hB, bl[L], (long long)N * HD, L < 2 ? 1 : 0);
    cur = hB;
    curK = HD;
  }

  // readout: mean over nodes, project with Wp, add bp
  gat_set_out<<<1, 32, 0, stream>>>((float*)d_out, bp);
  gat_readout<<<HD, 256, 0, stream>>>(hB, Wp, (float*)d_out, N);
}